// MambaOriginal_50173807952480
// MI455X (gfx1250) — compile-verified
//
#include <hip/hip_runtime.h>
#include <hip/hip_bf16.h>
#include <math.h>

// ---- problem constants (match reference) ----
#define B_      4
#define L_      512
#define DM      256          // D_MODEL
#define DI      512          // D_INNER
#define DS      16           // D_STATE
#define DCONV   4
#define DTR     16           // DT_RANK
#define NLAYERS 6
#define ROWS    (B_ * L_)    // 2048 token rows

typedef __attribute__((ext_vector_type(2))) float v2f;
typedef __attribute__((ext_vector_type(8))) float v8f;

// -------------------------------------------------------------------------
// RMSNorm: one 256-thread block per token row (D_MODEL = 256)
// -------------------------------------------------------------------------
__global__ __launch_bounds__(DM) void rmsnorm_kernel(
    const float* __restrict__ x, const float* __restrict__ w,
    float* __restrict__ out)
{
    __shared__ float red[DM];
    const int row = blockIdx.x;
    const int t   = threadIdx.x;
    const float v = x[(size_t)row * DM + t];
    red[t] = v * v;
    __syncthreads();
    for (int s = DM / 2; s > 0; s >>= 1) {
        if (t < s) red[t] += red[t + s];
        __syncthreads();
    }
    const float rstd = rsqrtf(red[0] * (1.0f / DM) + 1e-5f);
    out[(size_t)row * DM + t] = v * rstd * w[t];
}

// -------------------------------------------------------------------------
// Generic fp32 WMMA GEMM:  C[m,n] = sum_k A[m, k] * W[n, k]  (+bias)
//   act == 1 : softplus applied
//   residual : added if non-null (same layout as C)
// One wave computes a (16*MT) x (16*NT) block: per k-step load MT A-frags +
// NT B-frags, issue MT*NT V_WMMA_F32_16X16X4_F32 ops (2D register reuse).
// A-frag: lane l<16 -> row m0+l, K {k,k+1}; lane>=16 -> K {k+2,k+3}.
// B-frag: same pattern on W rows (B(k,n) = W[n][k]).
// C/D: vgpr v -> row m0 + 8*(lane/16) + v, col n0 + (lane&15).
// -------------------------------------------------------------------------
template <int MT, int NT>
__global__ __launch_bounds__(128) void gemm_wmma_f32(
    const float* __restrict__ A, int lda,
    const float* __restrict__ W,
    const float* __restrict__ bias,
    const float* __restrict__ residual,
    float* __restrict__ C,
    int N, int K, int ngroups, int act)
{
    const int wave = blockIdx.x * 4 + (threadIdx.x >> 5);
    const int m0   = (wave / ngroups) * (16 * MT);
    const int n0   = (wave % ngroups) * (16 * NT);
    const int lane = threadIdx.x & 31;
    const int half = lane >> 4;
    const int l16  = lane & 15;

    const float* aP = A + (size_t)(m0 + l16) * lda + half * 2;
    const float* wP = W + (size_t)(n0 + l16) * K   + half * 2;

    v8f acc[MT][NT];
#pragma unroll
    for (int i = 0; i < MT; ++i)
#pragma unroll
        for (int j = 0; j < NT; ++j)
            acc[i][j] = (v8f){0.f, 0.f, 0.f, 0.f, 0.f, 0.f, 0.f, 0.f};

    for (int k = 0; k < K; k += 4) {
        v2f av[MT];
#pragma unroll
        for (int i = 0; i < MT; ++i)
            av[i] = *(const v2f*)(aP + (size_t)i * 16 * lda + k);
        v2f bv[NT];
#pragma unroll
        for (int j = 0; j < NT; ++j)
            bv[j] = *(const v2f*)(wP + (size_t)j * 16 * K + k);
#pragma unroll
        for (int i = 0; i < MT; ++i)
#pragma unroll
            for (int j = 0; j < NT; ++j)
                acc[i][j] = __builtin_amdgcn_wmma_f32_16x16x4_f32(
                    /*neg_a=*/false, av[i], /*neg_b=*/false, bv[j],
                    /*c_mod=*/(short)0, acc[i][j],
                    /*reuse_a=*/false, /*reuse_b=*/false);
    }

#pragma unroll
    for (int i = 0; i < MT; ++i) {
        const int rowb = m0 + i * 16 + half * 8;
#pragma unroll
        for (int j = 0; j < NT; ++j) {
            const int   col  = n0 + j * 16 + l16;
            const float bval = bias ? bias[col] : 0.0f;
#pragma unroll
            for (int v = 0; v < 8; ++v) {
                float val = acc[i][j][v] + bval;
                if (act == 1) val = (val > 20.0f) ? val : log1pf(expf(val));
                const size_t off = (size_t)(rowb + v) * N + col;
                if (residual) val += residual[off];
                C[off] = val;
            }
        }
    }
}

// -------------------------------------------------------------------------
// Depthwise causal conv (taps=4) over xin = xz[..., :DI], then SiLU -> u
// one thread per (b, t, c)
// -------------------------------------------------------------------------
__global__ __launch_bounds__(256) void conv_silu_kernel(
    const float* __restrict__ xz, const float* __restrict__ cw,
    float* __restrict__ u)
{
    const int idx = blockIdx.x * blockDim.x + threadIdx.x;  // (b*L + t)*DI + c
    const int c  = idx & (DI - 1);
    const int bl = idx >> 9;            // b*L + t
    const int t  = bl & (L_ - 1);
    float acc = 0.0f;
#pragma unroll
    for (int j = 0; j < DCONV; ++j) {
        const int tt = t - (DCONV - 1) + j;
        if (tt >= 0)
            acc += xz[(size_t)(bl - (DCONV - 1) + j) * (2 * DI) + c] * cw[c * DCONV + j];
    }
    u[idx] = acc / (1.0f + expf(-acc));   // silu
}

// -------------------------------------------------------------------------
// Selective scan. Lane layout per wave32: lanes 0..15 -> channel d0 states
// n=0..15, lanes 16..31 -> channel d1. Sequential over t; h kept in a reg.
// y_final = (sum_n h*C_n + u*D) * silu(z), written by lane n==0 of each half.
// -------------------------------------------------------------------------
__global__ __launch_bounds__(256) void scan_kernel(
    const float* __restrict__ xdbc,   // (ROWS, 48): [dt | B | C]
    const float* __restrict__ delta,  // (ROWS, DI)
    const float* __restrict__ u,      // (ROWS, DI)
    const float* __restrict__ xz,     // (ROWS, 2*DI), z at cols [DI, 2*DI)
    const float* __restrict__ A_log,  // (DI, DS)
    const float* __restrict__ Dp,     // (DI)
    float* __restrict__ y)            // (ROWS, DI)
{
    const int gwave = (blockIdx.x * blockDim.x + threadIdx.x) >> 5;  // 0..1023
    const int lane  = threadIdx.x & 31;
    const int half  = lane >> 4;
    const int n     = lane & 15;
    const int b     = gwave >> 8;                 // 1024 waves / 256 per batch
    const int d     = ((gwave & 255) << 1) + half;

    const float Adn = -expf(A_log[d * DS + n]);
    const float Dd  = Dp[d];
    float h = 0.0f;

    for (int t = 0; t < L_; ++t) {
        const int row = b * L_ + t;
        const float dlt = delta[(size_t)row * DI + d];
        const float ut  = u[(size_t)row * DI + d];
        const float Bn  = xdbc[(size_t)row * 48 + DTR + n];
        const float Cn  = xdbc[(size_t)row * 48 + DTR + DS + n];

        h = expf(dlt * Adn) * h + (dlt * ut) * Bn;

        float yp = h * Cn;
        yp += __shfl_xor(yp, 1);
        yp += __shfl_xor(yp, 2);
        yp += __shfl_xor(yp, 4);
        yp += __shfl_xor(yp, 8);        // masks < 16: stays inside half

        if (n == 0) {
            const float z = xz[(size_t)row * (2 * DI) + DI + d];
            const float sz = z / (1.0f + expf(-z));
            y[(size_t)row * DI + d] = (yp + ut * Dd) * sz;
        }
    }
}

// -------------------------------------------------------------------------
// Host-side orchestration
// -------------------------------------------------------------------------
extern "C" void kernel_launch(void* const* d_in, const int* in_sizes, int n_in,
                              void* d_out, int out_size, void* d_ws, size_t ws_size,
                              hipStream_t stream)
{
    (void)in_sizes; (void)n_in; (void)out_size; (void)ws_size;

    const float* norm_w = (const float*)d_in[1];
    const float* W_in   = (const float*)d_in[2];
    const float* b_in   = (const float*)d_in[3];
    const float* conv_w = (const float*)d_in[4];
    const float* W_x    = (const float*)d_in[5];
    const float* dt_w   = (const float*)d_in[6];
    const float* dt_b   = (const float*)d_in[7];
    const float* A_log  = (const float*)d_in[8];
    const float* Dp     = (const float*)d_in[9];
    const float* W_out  = (const float*)d_in[10];
    const float* b_out  = (const float*)d_in[11];

    float* x = (float*)d_out;                    // live residual stream
    float* ws = (float*)d_ws;
    float* xn    = ws;                           // ROWS*DM     = 524288
    float* xz    = xn    + (size_t)ROWS * DM;    // ROWS*2*DI   = 2097152
    float* u     = xz    + (size_t)ROWS * 2 * DI;// ROWS*DI     = 1048576
    float* xdbc  = u     + (size_t)ROWS * DI;    // ROWS*48     = 98304
    float* delta = xdbc  + (size_t)ROWS * 48;    // ROWS*DI
    float* yb    = delta + (size_t)ROWS * DI;    // ROWS*DI

    // x <- input
    hipMemcpyAsync(x, d_in[0], (size_t)ROWS * DM * sizeof(float),
                   hipMemcpyDeviceToDevice, stream);

    // M tile-groups with MT=2: 2048/32 = 64
    const int MG = ROWS / 32;

    for (int i = 0; i < NLAYERS; ++i) {
        const float* nw  = norm_w + (size_t)i * DM;
        const float* Wi  = W_in   + (size_t)i * (2 * DI) * DM;
        const float* bi  = b_in   + (size_t)i * (2 * DI);
        const float* cw  = conv_w + (size_t)i * DI * DCONV;
        const float* Wx  = W_x    + (size_t)i * 48 * DI;
        const float* dw  = dt_w   + (size_t)i * DI * DTR;
        const float* db  = dt_b   + (size_t)i * DI;
        const float* Al  = A_log  + (size_t)i * DI * DS;
        const float* Di  = Dp     + (size_t)i * DI;
        const float* Wo  = W_out  + (size_t)i * DM * DI;
        const float* bo  = b_out  + (size_t)i * DM;

        // 1) xn = rmsnorm(x)
        rmsnorm_kernel<<<ROWS, DM, 0, stream>>>(x, nw, xn);

        // 2) xz = xn @ W_in^T + b_in   (M=2048, N=1024, K=256)
        //    2x4 tiles/wave -> 16 n-groups, 1024 waves, 256 blocks
        gemm_wmma_f32<2, 4><<<MG * (2 * DI / 64) / 4, 128, 0, stream>>>(
            xn, DM, Wi, bi, nullptr, xz, 2 * DI, DM, 2 * DI / 64, 0);

        // 3) u = silu(causal_conv(xz[...,:DI]))
        conv_silu_kernel<<<(ROWS * DI) / 256, 256, 0, stream>>>(xz, cw, u);

        // 4) xdbc = u @ W_x^T          (M=2048, N=48, K=512)
        //    2x3 tiles/wave -> one n-group of 48; 64 waves, 16 blocks
        gemm_wmma_f32<2, 3><<<MG / 4, 128, 0, stream>>>(
            u, DI, Wx, nullptr, nullptr, xdbc, 48, DI, 1, 0);

        // 5) delta = softplus(dt @ dt_w^T + dt_b)  (M=2048, N=512, K=16)
        //    dt is xdbc[:, 0:16] -> lda = 48; 8 n-groups, 512 waves, 128 blocks
        gemm_wmma_f32<2, 4><<<MG * (DI / 64) / 4, 128, 0, stream>>>(
            xdbc, 48, dw, db, nullptr, delta, DI, DTR, DI / 64, 1);

        // 6) selective scan + (+u*D) * silu(z) -> yb
        scan_kernel<<<(ROWS / 2) * 32 / 256, 256, 0, stream>>>(
            xdbc, delta, u, xz, Al, Di, yb);

        // 7) x = x + yb @ W_out^T + b_out   (M=2048, N=256, K=512)
        //    4 n-groups, 256 waves, 64 blocks
        gemm_wmma_f32<2, 4><<<MG * (DM / 64) / 4, 128, 0, stream>>>(
            yb, DI, Wo, bo, /*residual=*/x, x, DM, DI, DM / 64, 0);
    }
}